// PointIntegrator_63995012711113
// MI455X (gfx1250) — compile-verified
//
#include <hip/hip_runtime.h>

// NeRF-style ray compositing for MI455X (gfx1250, wave32).
//
// Roofline: reads 576 MB (colors 512 MB dominate), writes ~42 MB -> ~26 us at
// 23.3 TB/s HBM. Memory bound; fp32 throughout. The 16-channel weighted color
// reduction per ray is a (1x63)x(63x16) product mapped onto
// V_WMMA_F32_16X16X4_F32 (exact fp32 matrix op) with the weight chunk
// broadcast across the 16 M-rows; the transmittance cumprod is a wave-parallel
// prefix product via lane shuffles (ds_bpermute), co-executing with the XDL
// WMMA pipe.

typedef __attribute__((ext_vector_type(2))) float v2f;
typedef __attribute__((ext_vector_type(8))) float v8f;

#define NRAYS (2 * 65536)
#define KS    64
#define CCH   16
#define T_EPSF 1e-10f

// ---------------------------------------------------------------- init ws ---
__global__ void PI_init_ws(unsigned* ws) {
    ws[0] = 0x7F800000u;  // +inf  (running min of radii_m, as positive-float bits)
    ws[1] = 0x00000000u;  // 0.0f  (running max)
}

// ------------------------------------------------- global min/max radii_m ---
// radii are sorted along K per ray, so per-ray min/max midpoints are the
// first/last midpoints. 4 loads per ray, wave reduce, 1 atomic per wave.
__global__ __launch_bounds__(256) void PI_radii_minmax(
    const float* __restrict__ radii, unsigned* __restrict__ ws) {
    int ray = blockIdx.x * blockDim.x + threadIdx.x;
    float mn = __builtin_inff();
    float mx = 0.0f;
    if (ray < NRAYS) {
        const float* r = radii + (size_t)ray * KS;
        float2 lo = ((const float2*)r)[0];        // r[0], r[1]
        float2 hi = ((const float2*)r)[31];       // r[62], r[63]
        mn = 0.5f * (lo.x + lo.y);
        mx = 0.5f * (hi.x + hi.y);
    }
    for (int i = 16; i >= 1; i >>= 1) {
        mn = fminf(mn, __shfl_xor(mn, i, 32));
        mx = fmaxf(mx, __shfl_xor(mx, i, 32));
    }
    if ((threadIdx.x & 31) == 0) {
        atomicMin(&ws[0], __float_as_uint(mn));   // positive floats order as uints
        atomicMax(&ws[1], __float_as_uint(mx));
    }
}

// ------------------------------------------------------------- main pass ----
// One wave per ray. Lane l owns samples {2l, 2l+1} and intervals {2l, 2l+1}.
__global__ __launch_bounds__(256) void PI_integrate(
    const float* __restrict__ colors, const float* __restrict__ dens,
    const float* __restrict__ radii, const unsigned* __restrict__ ws,
    float* __restrict__ out_color, float* __restrict__ out_rad,
    float* __restrict__ out_w, float* __restrict__ out_T) {
    const int lane = threadIdx.x & 31;
    const int ray  = blockIdx.x * 8 + (threadIdx.x >> 5);

    const float* dR = dens   + (size_t)ray * KS;
    const float* rR = radii  + (size_t)ray * KS;
    const float* cR = colors + (size_t)ray * KS * CCH;

    // Samples: lane l loads k0=2l and k0+1 as one b64 load.
    const int k0 = lane << 1;
    float2 dd = ((const float2*)dR)[lane];
    float2 rr = ((const float2*)rR)[lane];
    float d0 = dd.x, d1 = dd.y;
    float r0 = rr.x, r1 = rr.y;
    // sample k0+2 comes from the next lane's first element
    float dn = __shfl_down(d0, 1, 32);
    float rn = __shfl_down(r0, 1, 32);

    // Interval j = 2l
    float dm0    = fmaxf(0.5f * (d0 + d1), 0.0f);
    float alpha0 = 1.0f - __expf(-(r1 - r0) * dm0);
    float rm0    = 0.5f * (r0 + r1);
    // Interval j = 2l+1 (invalid for lane 31: weight forced to 0)
    float dm1    = fmaxf(0.5f * (d1 + dn), 0.0f);
    float alpha1 = (lane < 31) ? (1.0f - __expf(-(rn - r1) * dm1)) : 0.0f;
    float rm1    = 0.5f * (r1 + rn);

    float g0 = 1.0f - alpha0 + T_EPSF;
    float g1 = (lane < 31) ? (1.0f - alpha1 + T_EPSF) : 1.0f;

    // Exclusive prefix product of g over 63 intervals (Hillis-Steele on pairs)
    float S = g0 * g1;
    for (int s = 1; s < 32; s <<= 1) {
        float t = __shfl_up(S, s, 32);
        if (lane >= s) S *= t;
    }
    float E = __shfl_up(S, 1, 32);          // exclusive lane prefix
    if (lane == 0) E = 1.0f;
    float P0 = E;                           // T[2l]
    float P1 = E * g0;                      // T[2l+1]
    float w0 = alpha0 * P0;
    float w1 = alpha1 * P1;

    // weights output (63 per ray)
    float* wout = out_w + (size_t)ray * (KS - 1);
    wout[k0] = w0;                          // k0 = 0..62 all valid
    if (lane < 31) wout[k0 + 1] = w1;

    // wave reductions: weight sum and radial-distance numerator
    float wsum = w0 + w1;
    float rsum = w0 * rm0 + w1 * rm1;
    for (int i = 16; i >= 1; i >>= 1) {
        wsum += __shfl_xor(wsum, i, 32);
        rsum += __shfl_xor(rsum, i, 32);
    }

    // Composite color via 16 chained V_WMMA_F32_16X16X4_F32 ops.
    // A(16x4): lanes 0-15 hold K=0/1, lanes 16-31 hold K=2/3 (ISA layout);
    // weights broadcast across M so every D row equals the composite color.
    // B(4x16): K-rows of color midpoints, N = lane&15.
    const int half = lane >> 4;             // which K-pair this lane supplies
    const int ch   = lane & 15;             // output channel
    v8f acc = {0.f, 0.f, 0.f, 0.f, 0.f, 0.f, 0.f, 0.f};
    for (int c = 0; c < 16; ++c) {
        v2f a;
        a.x = __shfl(w0, 2 * c + half, 32); // w[4c + 2*half]      (even j)
        a.y = __shfl(w1, 2 * c + half, 32); // w[4c + 2*half + 1]  (odd j)

        int jx = 4 * c + 2 * half;          // first interval row for this lane
        float cA = cR[jx * CCH + ch];
        float cB = cR[(jx + 1) * CCH + ch];
        int j2 = (jx + 2 > 63) ? 63 : (jx + 2);  // clamp: interval 63 has w=0
        float cC = cR[j2 * CCH + ch];
        v2f b;
        b.x = 0.5f * (cA + cB);             // colors_m[jx][ch]
        b.y = 0.5f * (cB + cC);             // colors_m[jx+1][ch]

        acc = __builtin_amdgcn_wmma_f32_16x16x4_f32(
            /*neg_a=*/false, a, /*neg_b=*/false, b,
            /*c_mod=*/(short)0, acc, /*reuse_a=*/false, /*reuse_b=*/false);
    }

    // Remaining scalars
    float t_end = __shfl(P0, 31, 32);       // T[62] = prod_{i<62} g_i
    float gmin  = __uint_as_float(ws[0]);
    float gmax  = __uint_as_float(ws[1]);
    float rad   = rsum / wsum;
    if (rad != rad) rad = __builtin_inff(); // nan_to_num(nan -> +inf)
    rad = fminf(fmaxf(rad, gmin), gmax);    // clip to global [min,max] radii_m

    if (lane < 16)
        out_color[(size_t)ray * CCH + lane] = acc[0] * 2.0f - 1.0f; // D row M=0
    if (lane == 0) {
        out_rad[ray] = rad;
        out_T[ray]   = t_end;
    }
}

// ------------------------------------------------------------------ launch --
extern "C" void kernel_launch(void* const* d_in, const int* in_sizes, int n_in,
                              void* d_out, int out_size, void* d_ws, size_t ws_size,
                              hipStream_t stream) {
    const float* colors = (const float*)d_in[0];
    const float* dens   = (const float*)d_in[1];
    const float* radii  = (const float*)d_in[2];

    float* out = (float*)d_out;
    // outputs concatenated flat in return order
    float* out_color = out;                                 // N*R*C  = 2,097,152
    float* out_rad   = out + (size_t)NRAYS * CCH;           // N*R    =   131,072
    float* out_w     = out_rad + NRAYS;                     // N*R*63 = 8,257,536
    float* out_T     = out_w + (size_t)NRAYS * (KS - 1);    // N*R    =   131,072

    unsigned* ws = (unsigned*)d_ws;

    PI_init_ws<<<1, 1, 0, stream>>>(ws);
    PI_radii_minmax<<<(NRAYS + 255) / 256, 256, 0, stream>>>(radii, ws);
    PI_integrate<<<NRAYS / 8, 256, 0, stream>>>(colors, dens, radii, ws,
                                                out_color, out_rad, out_w, out_T);
}